// ConceptNet_34162169873048
// MI455X (gfx1250) — compile-verified
//
#include <hip/hip_runtime.h>
#include <hip/hip_bf16.h>

typedef __attribute__((ext_vector_type(2))) float v2f;
typedef __attribute__((ext_vector_type(4))) float v4f;
typedef __attribute__((ext_vector_type(8))) float v8f;

#define BS   8192
#define D    768
#define NC   64
#define NCL  500
#define CS   100
#define NY   1000
#define EPS  1e-12f

// ---------------------------------------------------------------- cluster mean
// clusters [500,100,768] -> mean [500,768].  153 MB stream; b128 loads,
// 192 threads * 16B = full 3KB row per block iteration, fully coalesced.
__global__ void kmean(const float* __restrict__ clusters, float* __restrict__ mean) {
    int c  = blockIdx.x;
    int d4 = threadIdx.x;                         // 192 threads = 768/4
    const v4f* p = (const v4f*)(clusters + (size_t)c * CS * D) + d4;
    v4f s = {};
    #pragma unroll 4
    for (int j = 0; j < CS; ++j) {
        v4f v = p[(size_t)j * (D / 4)];
        s.x += v.x; s.y += v.y; s.z += v.z; s.w += v.w;
    }
    const float inv = 1.0f / CS;
    s.x *= inv; s.y *= inv; s.z *= inv; s.w *= inv;
    ((v4f*)(mean + (size_t)c * D))[d4] = s;
}

// ---------------------------------------------------------------- column norms of concept
__global__ void kcnorm(const float* __restrict__ C, float* __restrict__ cnorm) {
    int j = threadIdx.x;                          // 64 threads
    float s = 0.f;
    for (int k = 0; k < D; ++k) { float v = C[(size_t)k * NC + j]; s += v * v; }
    cnorm[j] = sqrtf(s);
}

// ---------------------------------------------------------------- gram = C^T C  [64,64]
__global__ void kgram(const float* __restrict__ C, float* __restrict__ G) {
    int idx = blockIdx.x * 256 + threadIdx.x;     // 16 blocks * 256 = 4096
    int i = idx >> 6, j = idx & 63;
    float s = 0.f;
    for (int k = 0; k < D; ++k) s += C[(size_t)k * NC + i] * C[(size_t)k * NC + j];
    G[idx] = s;
}

// ---------------------------------------------------------------- 64x64 SPD inverse (Gauss-Jordan in LDS)
__global__ void kinv(const float* __restrict__ G, float* __restrict__ Minv) {
    __shared__ float A[64][128];                  // 32 KB augmented [G | I]
    __shared__ float fac[64];
    int tid = threadIdx.x;                        // 256 threads
    for (int t = tid; t < 64 * 128; t += 256) {
        int r = t >> 7, c = t & 127;
        A[r][c] = (c < 64) ? G[r * 64 + c] : ((c - 64 == r) ? 1.0f : 0.0f);
    }
    __syncthreads();
    for (int k = 0; k < 64; ++k) {
        float pivinv = 1.0f / A[k][k];
        if (tid < 64) fac[tid] = A[tid][k];
        __syncthreads();
        for (int c = tid; c < 128; c += 256) A[k][c] *= pivinv;
        __syncthreads();
        for (int t = tid; t < 64 * 128; t += 256) {
            int r = t >> 7, c = t & 127;
            if (r != k) A[r][c] -= fac[r] * A[k][c];
        }
        __syncthreads();
    }
    for (int t = tid; t < 64 * 64; t += 256) {
        int r = t >> 6, c = t & 63;
        Minv[t] = A[r][64 + c];
    }
}

// ---------------------------------------------------------------- S = |mean @ C| / colnorm   [500,64]
__global__ void kscore(const float* __restrict__ mean, const float* __restrict__ C,
                       const float* __restrict__ cnorm, float* __restrict__ S) {
    int c = blockIdx.x, j = threadIdx.x;          // 500 blocks x 64 threads
    const float* m = mean + (size_t)c * D;
    float s = 0.f;
    #pragma unroll 4
    for (int k = 0; k < D; ++k) s += m[k] * C[(size_t)k * NC + j];
    S[(size_t)c * NC + j] = fabsf(s) / fmaxf(cnorm[j], EPS);
}

// ---------------------------------------------------------------- sparsity losses (scalars)
// L1 = sum(Sn);  L2 = ||rowsum(Sn)||^2 - trace(Sn^T Sn)
__global__ void kloss(const float* __restrict__ S, float* __restrict__ out_scalars) {
    __shared__ float colinv[64];
    __shared__ float col2[64];
    __shared__ float red[256];
    __shared__ float L1s;
    int tid = threadIdx.x;                        // 256 threads
    {   // column sum of squares over 500 rows
        int j = tid & 63, seg = tid >> 6;         // 4 segments
        float s = 0.f;
        for (int c = seg; c < NCL; c += 4) { float v = S[(size_t)c * NC + j]; s += v * v; }
        red[tid] = s;
        __syncthreads();
        if (tid < 64) {
            float t = red[tid] + red[tid + 64] + red[tid + 128] + red[tid + 192];
            col2[tid] = t;
            colinv[tid] = 1.0f / fmaxf(sqrtf(t), EPS);
        }
        __syncthreads();
    }
    float a1 = 0.f, a2 = 0.f;
    for (int c = tid; c < NCL; c += 256) {
        float r = 0.f;
        #pragma unroll
        for (int j = 0; j < 64; ++j) r += S[(size_t)c * NC + j] * colinv[j];
        a1 += r; a2 += r * r;
    }
    red[tid] = a1; __syncthreads();
    for (int s2 = 128; s2 > 0; s2 >>= 1) { if (tid < s2) red[tid] += red[tid + s2]; __syncthreads(); }
    if (tid == 0) L1s = red[0];
    __syncthreads();
    red[tid] = a2; __syncthreads();
    for (int s2 = 128; s2 > 0; s2 >>= 1) { if (tid < s2) red[tid] += red[tid + s2]; __syncthreads(); }
    if (tid == 0) {
        float tr = 0.f;
        for (int j = 0; j < 64; ++j) tr += col2[j] * colinv[j] * colinv[j];
        out_scalars[0] = L1s;
        out_scalars[1] = red[0] - tr;
    }
}

// ---------------------------------------------------------------- K1 = C^T @ W_h  [64,1000]
__global__ void kK1(const float* __restrict__ C, const float* __restrict__ W,
                    float* __restrict__ K1) {
    int n = blockIdx.x * 128 + threadIdx.x;       // 8 blocks
    int i = blockIdx.y;                           // 64
    if (n >= NY) return;
    float s = 0.f;
    #pragma unroll 4
    for (int k = 0; k < D; ++k) s += C[(size_t)k * NC + i] * W[(size_t)k * NY + n];
    K1[(size_t)i * NY + n] = s;
}

// ---------------------------------------------------------------- K2 = Minv @ K1  [64,1000]
__global__ void kK2(const float* __restrict__ Minv, const float* __restrict__ K1,
                    float* __restrict__ K2) {
    int n = blockIdx.x * 128 + threadIdx.x;
    int i = blockIdx.y;
    if (n >= NY) return;
    float s = 0.f;
    #pragma unroll
    for (int j = 0; j < 64; ++j) s += Minv[i * 64 + j] * K1[(size_t)j * NY + n];
    K2[(size_t)i * NY + n] = s;
}

// ---------------------------------------------------------------- GEMM1: T1 = E @ C  [8192,64]  (WMMA f32 16x16x4)
// One wave computes a 16x64 tile (4 N-tiles). 512 tiles / 4 waves per block = 128 blocks.
__global__ void gemm_EC(const float* __restrict__ E, const float* __restrict__ C,
                        float* __restrict__ T1) {
    int wave = threadIdx.x >> 5;
    int lane = threadIdx.x & 31;
    int mt = blockIdx.x * 4 + wave;               // 0..511
    int m0 = mt * 16;
    int row   = lane & 15;                        // A: M,  B: N within tile
    int khalf = lane >> 4;                        // selects K pair
    v8f a0 = {}, a1 = {}, a2 = {}, a3 = {};
    const float* Arow = E + (size_t)(m0 + row) * D;
    for (int k0 = 0; k0 < D; k0 += 4) {
        int ka = k0 + 2 * khalf;
        v2f av = *(const v2f*)(Arow + ka);        // K=ka, ka+1 (8B aligned)
        const float* B0 = C + (size_t)ka * NC;    // row ka of C
        const float* B1 = C + (size_t)(ka + 1) * NC;
        v2f b0 = { B0[ 0 + row], B1[ 0 + row] };
        v2f b1 = { B0[16 + row], B1[16 + row] };
        v2f b2 = { B0[32 + row], B1[32 + row] };
        v2f b3 = { B0[48 + row], B1[48 + row] };
        a0 = __builtin_amdgcn_wmma_f32_16x16x4_f32(false, av, false, b0, (short)0, a0, false, false);
        a1 = __builtin_amdgcn_wmma_f32_16x16x4_f32(false, av, false, b1, (short)0, a1, false, false);
        a2 = __builtin_amdgcn_wmma_f32_16x16x4_f32(false, av, false, b2, (short)0, a2, false, false);
        a3 = __builtin_amdgcn_wmma_f32_16x16x4_f32(false, av, false, b3, (short)0, a3, false, false);
    }
    int n = lane & 15;
    int mbase = (lane >> 4) * 8;
    float* out = T1 + (size_t)m0 * NC;
    #pragma unroll
    for (int r = 0; r < 8; ++r) {
        size_t ro = (size_t)(mbase + r) * NC;
        out[ro +  0 + n] = a0[r];
        out[ro + 16 + n] = a1[r];
        out[ro + 32 + n] = a2[r];
        out[ro + 48 + n] = a3[r];
    }
}

// ---------------------------------------------------------------- GEMM2: Y = T1 @ K2 + b  [8192,1000] (WMMA f32 16x16x4)
// One wave per 16x16 tile; 512 x 63 tiles = 32256 waves = 4032 blocks of 256.
__global__ void gemm_out(const float* __restrict__ T1, const float* __restrict__ K2,
                         const float* __restrict__ bias, float* __restrict__ Y) {
    int tile = blockIdx.x * 8 + (threadIdx.x >> 5);
    int mt = tile / 63, nt = tile % 63;
    int lane = threadIdx.x & 31;
    int row = lane & 15, khalf = lane >> 4;
    int m0 = mt * 16, n0 = nt * 16;
    int nb = n0 + row;                            // B column for this lane
    int nclamp = nb < NY ? nb : NY - 1;           // clamp loads in last tile
    const float* Arow = T1 + (size_t)(m0 + row) * NC;
    v8f acc = {};
    #pragma unroll
    for (int k0 = 0; k0 < NC; k0 += 4) {
        int ka = k0 + 2 * khalf;
        v2f av = *(const v2f*)(Arow + ka);
        v2f bv = { K2[(size_t)ka * NY + nclamp], K2[(size_t)(ka + 1) * NY + nclamp] };
        acc = __builtin_amdgcn_wmma_f32_16x16x4_f32(false, av, false, bv, (short)0, acc, false, false);
    }
    int n = n0 + (lane & 15);
    if (n < NY) {
        float bvv = bias[n];
        int mbase = (lane >> 4) * 8;
        #pragma unroll
        for (int r = 0; r < 8; ++r)
            Y[(size_t)(m0 + mbase + r) * NY + n] = acc[r] + bvv;
    }
}

// ---------------------------------------------------------------- launch
extern "C" void kernel_launch(void* const* d_in, const int* in_sizes, int n_in,
                              void* d_out, int out_size, void* d_ws, size_t ws_size,
                              hipStream_t stream) {
    const float* E        = (const float*)d_in[0];   // [8192,768]
    const float* clusters = (const float*)d_in[1];   // [500,100,768]
    const float* C        = (const float*)d_in[2];   // [768,64]
    const float* W_h      = (const float*)d_in[3];   // [768,1000]
    const float* b_h      = (const float*)d_in[4];   // [1000]
    float* out = (float*)d_out;                      // y_pred (8192000) + L1 + L2

    float* ws = (float*)d_ws;
    float* mean  = ws;                               // 384000
    float* S     = mean  + (size_t)NCL * D;          // 32000
    float* G     = S     + (size_t)NCL * NC;         // 4096
    float* Minv  = G     + 64 * 64;                  // 4096
    float* cnorm = Minv  + 64 * 64;                  // 64
    float* K1    = cnorm + 64;                       // 64000
    float* K2    = K1    + (size_t)NC * NY;          // 64000
    float* T1    = K2    + (size_t)NC * NY;          // 524288

    // losses path
    kmean <<<NCL, D / 4, 0, stream>>>(clusters, mean);
    kcnorm<<<1, 64, 0, stream>>>(C, cnorm);
    kscore<<<NCL, 64, 0, stream>>>(mean, C, cnorm, S);
    kloss <<<1, 256, 0, stream>>>(S, out + (size_t)BS * NY);

    // projection head path: y = (E C) inv(G) (C^T W) + b
    kgram <<<16, 256, 0, stream>>>(C, G);
    kinv  <<<1, 256, 0, stream>>>(G, Minv);
    kK1   <<<dim3(8, 64), 128, 0, stream>>>(C, W_h, K1);
    kK2   <<<dim3(8, 64), 128, 0, stream>>>(Minv, K1, K2);
    gemm_EC <<<128, 128, 0, stream>>>(E, C, T1);
    gemm_out<<<4032, 256, 0, stream>>>(T1, K2, b_h, out);
}